// RQCodebook_89799176224926
// MI455X (gfx1250) — compile-verified
//
#include <hip/hip_runtime.h>

#define K_CODES 4096
#define DIM     512
#define NQ      4
#define NROWS   (16 * 2048)   // 32768
#define BETA    0.25f
#define A_STRIDE 516          // 512 + 4: lanes 0-15 hit banks {4r,4r+1}, lanes 16-31 {4r+2,4r+3} -> conflict-free

typedef float v2f __attribute__((ext_vector_type(2)));
typedef float v8f __attribute__((ext_vector_type(8)));

// ---------------------------------------------------------------- init
__global__ void init_zero(float* __restrict__ zq, float* __restrict__ lossAcc) {
  size_t n = (size_t)NROWS * DIM;
  size_t stride = (size_t)gridDim.x * blockDim.x;
  for (size_t i = (size_t)blockIdx.x * blockDim.x + threadIdx.x; i < n; i += stride)
    zq[i] = 0.0f;
  if (blockIdx.x == 0 && threadIdx.x == 0) *lossAcc = 0.0f;
}

// ---------------------------------------------------------------- ||c||^2 for every code of every stage
__global__ __launch_bounds__(256) void cnorm_kernel(const float* __restrict__ cb,
                                                    float* __restrict__ cnorm) {
  int wid  = (int)((blockIdx.x * blockDim.x + threadIdx.x) >> 5); // one wave per code
  int lane = threadIdx.x & 31;
  if (wid >= NQ * K_CODES) return;
  const float* c = cb + (size_t)wid * DIM;
  float s = 0.0f;
  for (int k = lane; k < DIM; k += 32) { float v = c[k]; s += v * v; }
  #pragma unroll
  for (int off = 16; off; off >>= 1) s += __shfl_xor(s, off, 32);
  if (lane == 0) cnorm[wid] = s;
}

// ---------------------------------------------------------------- WMMA argmin over all codes
// Block = 256 threads (8 waves) handles 16 rows. Residual tile in LDS.
// Each wave scans 512 codes = 8 groups of 4 code-tiles; 4 independent
// v_wmma_f32_16x16x4_f32 accumulation chains per group, A-frag shared.
__global__ __launch_bounds__(256) void argmin_kernel(
    const float* __restrict__ z, const float* __restrict__ zqacc,
    const float* __restrict__ cb,      // this stage's codebook [K_CODES][DIM]
    const float* __restrict__ cnorm,   // this stage's ||c||^2  [K_CODES]
    int* __restrict__ idxOut)          // [NROWS] for this stage
{
  __shared__ float ldsA[16 * A_STRIDE];
  __shared__ float redv[8][16];
  __shared__ int   redi[8][16];

  const int tid  = threadIdx.x;
  const int row0 = blockIdx.x * 16;

  // stage residual tile: r = z - zq_accum
  {
    size_t gbase = (size_t)row0 * DIM;
    for (int i = tid; i < 16 * DIM; i += 256) {
      int r = i >> 9, k = i & (DIM - 1);
      size_t g = gbase + (size_t)i;
      ldsA[r * A_STRIDE + k] = z[g] - zqacc[g];
    }
  }
  __syncthreads();

  const int lane = tid & 31;
  const int wave = tid >> 5;
  const int col  = lane & 15;          // B column / A row (per ISA frag layouts)
  const int kh   = (lane >> 4) << 1;   // K sub-offset: 0 for lanes 0-15, 2 for 16-31
  const float* aBase = &ldsA[col * A_STRIDE + kh];

  float bestv[8];
  int   besti[8];
  #pragma unroll
  for (int v = 0; v < 8; ++v) { bestv[v] = 3.4e38f; besti[v] = 0; }

#define WMMA4(ACC, AV, BV)                                                 \
  (ACC) = __builtin_amdgcn_wmma_f32_16x16x4_f32(                           \
      false, (AV), false, (BV), (short)0, (ACC), false, false)

#define ARGMIN_TILE(ACC, TT)                                               \
  {                                                                        \
    const int  code = tile0 + (TT) * 16 + col;                             \
    const float cn2 = cnorm[code];                                         \
    _Pragma("unroll")                                                      \
    for (int v = 0; v < 8; ++v) {                                          \
      float s = cn2 - 2.0f * (ACC)[v];                                     \
      if (s < bestv[v]) { bestv[v] = s; besti[v] = code; }                 \
    }                                                                      \
  }

  const int code0 = wave * 512;        // this wave's code range
  for (int g = 0; g < 8; ++g) {
    const int tile0 = code0 + g * 64;  // 4 code tiles: tile0, +16, +32, +48
    const float* bp0 = cb + (size_t)(tile0 + col) * DIM + kh;
    const float* bp1 = bp0 + (size_t)16 * DIM;
    const float* bp2 = bp0 + (size_t)32 * DIM;
    const float* bp3 = bp0 + (size_t)48 * DIM;

    v8f acc0 = {}, acc1 = {}, acc2 = {}, acc3 = {};
    #pragma unroll 2
    for (int k0 = 0; k0 < DIM; k0 += 4) {
      v2f a  = *(const v2f*)(aBase + k0);   // A 16x4 frag: lane row, K=kh..kh+1
      v2f b0 = *(const v2f*)(bp0 + k0);     // B 4x16 frags: lane col, K=kh..kh+1
      v2f b1 = *(const v2f*)(bp1 + k0);
      v2f b2 = *(const v2f*)(bp2 + k0);
      v2f b3 = *(const v2f*)(bp3 + k0);
      WMMA4(acc0, a, b0);                   // 4 independent accumulation chains
      WMMA4(acc1, a, b1);
      WMMA4(acc2, a, b2);
      WMMA4(acc3, a, b3);
    }
    // score = ||c||^2 - 2 r.c  (||r||^2 constant per row, dropped);
    // tiles processed in increasing code order -> argmin-first tie-break.
    ARGMIN_TILE(acc0, 0);
    ARGMIN_TILE(acc1, 1);
    ARGMIN_TILE(acc2, 2);
    ARGMIN_TILE(acc3, 3);
  }
#undef WMMA4
#undef ARGMIN_TILE

  // reduce across the 16 columns held in each lane half; argmin-first tie-break
  #pragma unroll
  for (int v = 0; v < 8; ++v) {
    float bv = bestv[v];
    int   bi = besti[v];
    #pragma unroll
    for (int off = 1; off < 16; off <<= 1) {
      float ov = __shfl_xor(bv, off, 32);
      int   oi = __shfl_xor(bi, off, 32);
      if (ov < bv || (ov == bv && oi < bi)) { bv = ov; bi = oi; }
    }
    // C layout: lanes 0-15 -> row v ; lanes 16-31 -> row v+8
    if (lane == 0)  { redv[wave][v]     = bv; redi[wave][v]     = bi; }
    if (lane == 16) { redv[wave][v + 8] = bv; redi[wave][v + 8] = bi; }
  }
  __syncthreads();

  if (tid < 16) {
    float bv = redv[0][tid];
    int   bi = redi[0][tid];
    #pragma unroll
    for (int w = 1; w < 8; ++w) {
      float ov = redv[w][tid];
      int   oi = redi[w][tid];
      if (ov < bv || (ov == bv && oi < bi)) { bv = ov; bi = oi; }
    }
    idxOut[row0 + tid] = bi;
  }
}

// ---------------------------------------------------------------- gather + z_q accumulate + loss
__global__ __launch_bounds__(256) void update_kernel(
    const float* __restrict__ z, float* __restrict__ zqacc,
    const float* __restrict__ cb, const int* __restrict__ idx,
    float* __restrict__ lossAcc)
{
  __shared__ float wsum[8];
  const int row  = blockIdx.x;
  const int tid  = threadIdx.x;
  const int code = idx[row];
  const float* c = cb + (size_t)code * DIM;
  const size_t base = (size_t)row * DIM;

  float ssq = 0.0f;
  #pragma unroll
  for (int j = 0; j < 2; ++j) {
    int k = tid + j * 256;
    float zq   = c[k];
    float za   = zqacc[base + k];
    float rold = z[base + k] - za;       // residual before this stage
    float d    = 2.0f * zq - rold;       // zq - (rold - zq)
    ssq += d * d;
    zqacc[base + k] = za + zq;
  }
  #pragma unroll
  for (int off = 16; off; off >>= 1) ssq += __shfl_xor(ssq, off, 32);
  if ((tid & 31) == 0) wsum[tid >> 5] = ssq;
  __syncthreads();
  if (tid == 0) {
    float s = 0.0f;
    #pragma unroll
    for (int w = 0; w < 8; ++w) s += wsum[w];
    atomicAdd(lossAcc, s);
  }
}

// ---------------------------------------------------------------- finalize
__global__ void finalize_idx(const int* __restrict__ idxAll /*[NQ][NROWS]*/,
                             float* __restrict__ out) {
  int i = blockIdx.x * blockDim.x + threadIdx.x;
  if (i < NROWS * NQ) {
    int n = i >> 2, q = i & 3;                 // layout: (n, q) row-major
    out[i] = (float)idxAll[q * NROWS + n];
  }
}

__global__ void finalize_loss(const float* __restrict__ acc, float* __restrict__ out) {
  out[0] = (1.0f + BETA) * acc[0] / (float)((size_t)NROWS * DIM);
}

// ---------------------------------------------------------------- launch
extern "C" void kernel_launch(void* const* d_in, const int* in_sizes, int n_in,
                              void* d_out, int out_size, void* d_ws, size_t ws_size,
                              hipStream_t stream) {
  const float* z   = (const float*)d_in[0];                 // [B,L,DIM]
  const float* cbs = (const float*)d_in[1];                 // [NQ,K_CODES,DIM]

  float* out     = (float*)d_out;
  float* zq      = out;                                     // [NROWS*DIM]
  float* outIdx  = out + (size_t)NROWS * DIM;               // [NROWS*NQ]
  float* outLoss = outIdx + (size_t)NROWS * NQ;             // [1]

  char*  ws      = (char*)d_ws;
  float* cnorm   = (float*)ws;                              // NQ*K_CODES floats
  int*   idxAll  = (int*)(ws + (size_t)NQ * K_CODES * sizeof(float)); // NQ*NROWS ints
  float* lossAcc = (float*)(ws + (size_t)NQ * K_CODES * sizeof(float)
                               + (size_t)NQ * NROWS * sizeof(int));   // 1 float

  init_zero<<<1024, 256, 0, stream>>>(zq, lossAcc);
  cnorm_kernel<<<(NQ * K_CODES) / 8, 256, 0, stream>>>(cbs, cnorm);

  for (int s = 0; s < NQ; ++s) {
    const float* cb = cbs + (size_t)s * K_CODES * DIM;
    argmin_kernel<<<NROWS / 16, 256, 0, stream>>>(
        z, zq, cb, cnorm + (size_t)s * K_CODES, idxAll + (size_t)s * NROWS);
    update_kernel<<<NROWS, 256, 0, stream>>>(
        z, zq, cb, idxAll + (size_t)s * NROWS, lossAcc);
  }

  finalize_idx<<<(NROWS * NQ + 255) / 256, 256, 0, stream>>>(idxAll, outIdx);
  finalize_loss<<<1, 1, 0, stream>>>(lossAcc, outLoss);
}